// RegressionModel_10359461118193
// MI455X (gfx1250) — compile-verified
//
#include <hip/hip_runtime.h>
#include <math.h>

// ---------------- CDNA5 WMMA types ----------------
typedef __attribute__((ext_vector_type(2))) float v2f;
typedef __attribute__((ext_vector_type(8))) float v8f;

#define NEG_SLOPE 0.2f

// Monotone float -> uint mapping so unsigned atomicMax == float max.
__device__ __forceinline__ unsigned fmap(float f) {
    unsigned u = __float_as_uint(f);
    return (u & 0x80000000u) ? ~u : (u | 0x80000000u);
}
__device__ __forceinline__ float funmap(unsigned u) {
    unsigned v = (u & 0x80000000u) ? (u & 0x7fffffffu) : ~u;
    return __uint_as_float(v);
}

// ---------------- Layer 1 feature transform: h1 = x @ W1 ----------------
// One wave32 per 16-node tile. K=128 via 32 chained V_WMMA_F32_16X16X4_F32.
__global__ void k_gemm1(const float* __restrict__ x, const float* __restrict__ W1,
                        float* __restrict__ h1, int N) {
    __shared__ float Wl[128 * 16];
    for (int t = threadIdx.x; t < 128 * 16; t += blockDim.x) Wl[t] = W1[t];
    __syncthreads();

    int wave = (blockIdx.x * blockDim.x + threadIdx.x) >> 5;
    int lane = threadIdx.x & 31;
    int base = wave * 16;
    if (base >= N) return;                 // uniform per wave: EXEC stays all-1s
    int m = lane & 15, half = lane >> 4;

    const float* arow = x + (size_t)(base + m) * 128 + 2 * half;
    v8f c = {};
#pragma unroll
    for (int k = 0; k < 128; k += 4) {
        v2f a = *(const v2f*)(arow + k);   // A[M=m][K=k+2h, k+2h+1]
        int kk = k + 2 * half;
        v2f b;
        b.x = Wl[kk * 16 + m];             // B[K=kk][N=m]
        b.y = Wl[kk * 16 + 16 + m];        // B[K=kk+1][N=m]
        c = __builtin_amdgcn_wmma_f32_16x16x4_f32(false, a, false, b,
                                                  (short)0, c, false, false);
    }
#pragma unroll
    for (int r = 0; r < 8; ++r) {
        int node = base + r + 8 * half;    // D layout: M = r + 8*half, N = m
        h1[(size_t)node * 16 + m] = c[r];
    }
}

// ---------------- Per-node attention logits + buffer init (layer 1) -------
__global__ void k_prep1(const float* __restrict__ h1, const float* __restrict__ a_src,
                        const float* __restrict__ a_dst, float* __restrict__ asrc,
                        float* __restrict__ adst, unsigned* __restrict__ mx,
                        float* __restrict__ denom, float* __restrict__ agg, int N) {
    int i = blockIdx.x * blockDim.x + threadIdx.x;
    if (i >= N) return;
    const float* hr = h1 + (size_t)i * 16;
    float s = 0.f, d = 0.f;
#pragma unroll
    for (int j = 0; j < 16; ++j) { float v = hr[j]; s += v * a_src[j]; d += v * a_dst[j]; }
    asrc[i] = s; adst[i] = d;
    mx[i] = 0u;                   // below fmap of any finite float
    denom[i] = 0.f;
    float* ar = agg + (size_t)i * 16;
#pragma unroll
    for (int j = 0; j < 16; ++j) ar[j] = 0.f;
}

// ---------------- Edge pass 1: per-dst max of leaky_relu(e) --------------
__global__ void k_edge_max(const int* __restrict__ src, const int* __restrict__ dst,
                           const float* __restrict__ asrc, const float* __restrict__ adst,
                           unsigned* __restrict__ mx, int E, int N) {
    int t = blockIdx.x * blockDim.x + threadIdx.x;
    if (t >= E + N) return;
    int s, d;
    if (t < E) { s = src[t]; d = dst[t]; } else { s = t - E; d = s; }  // self loop
    float e = asrc[s] + adst[d];
    e = e > 0.f ? e : NEG_SLOPE * e;
    atomicMax(&mx[d], fmap(e));
}

// ---------------- Edge pass 2: per-dst sum of exp(e - max) ---------------
__global__ void k_edge_sum(const int* __restrict__ src, const int* __restrict__ dst,
                           const float* __restrict__ asrc, const float* __restrict__ adst,
                           const unsigned* __restrict__ mx, float* __restrict__ denom,
                           int E, int N) {
    int t = blockIdx.x * blockDim.x + threadIdx.x;
    if (t >= E + N) return;
    int s, d;
    if (t < E) { s = src[t]; d = dst[t]; } else { s = t - E; d = s; }
    float e = asrc[s] + adst[d];
    e = e > 0.f ? e : NEG_SLOPE * e;
    atomicAdd(&denom[d], expf(e - funmap(mx[d])));
}

// ---------------- Edge pass 3: agg[d] += alpha * h[s] --------------------
template <int D>
__global__ void k_edge_agg(const int* __restrict__ src, const int* __restrict__ dst,
                           const float* __restrict__ asrc, const float* __restrict__ adst,
                           const unsigned* __restrict__ mx, const float* __restrict__ denom,
                           const float* __restrict__ h, float* __restrict__ agg,
                           int E, int N) {
    int t = blockIdx.x * blockDim.x + threadIdx.x;
    if (t >= E + N) return;
    int s, d;
    if (t < E) { s = src[t]; d = dst[t]; } else { s = t - E; d = s; }
    float e = asrc[s] + adst[d];
    e = e > 0.f ? e : NEG_SLOPE * e;
    float alpha = expf(e - funmap(mx[d])) / denom[d];
    const float* hr = h + (size_t)s * 16;    // rows padded to 16 floats
    float* ar = agg + (size_t)d * 16;
#pragma unroll
    for (int j = 0; j < D; ++j) atomicAdd(&ar[j], alpha * hr[j]);
}

// --------- Layer 2 transform: h2 = relu(agg1 + b1) @ W2 (16->10) ---------
// bias+ReLU fused into the A operand; W2 zero-padded 10->16 cols in LDS.
__global__ void k_gemm2(const float* __restrict__ agg1, const float* __restrict__ b1,
                        const float* __restrict__ W2, float* __restrict__ h2, int N) {
    __shared__ float Wl[16 * 16];
    for (int t = threadIdx.x; t < 256; t += blockDim.x) {
        int k = t >> 4, n = t & 15;
        Wl[t] = (n < 10) ? W2[k * 10 + n] : 0.f;
    }
    __syncthreads();

    int wave = (blockIdx.x * blockDim.x + threadIdx.x) >> 5;
    int lane = threadIdx.x & 31;
    int base = wave * 16;
    if (base >= N) return;
    int m = lane & 15, half = lane >> 4;

    const float* arow = agg1 + (size_t)(base + m) * 16 + 2 * half;
    v8f c = {};
#pragma unroll
    for (int k = 0; k < 16; k += 4) {
        int kk = k + 2 * half;
        v2f av = *(const v2f*)(arow + k);
        v2f a, b;
        a.x = fmaxf(av.x + b1[kk], 0.f);       // relu(gat1 output + bias)
        a.y = fmaxf(av.y + b1[kk + 1], 0.f);
        b.x = Wl[kk * 16 + m];
        b.y = Wl[kk * 16 + 16 + m];
        c = __builtin_amdgcn_wmma_f32_16x16x4_f32(false, a, false, b,
                                                  (short)0, c, false, false);
    }
#pragma unroll
    for (int r = 0; r < 8; ++r) {
        int node = base + r + 8 * half;
        h2[(size_t)node * 16 + m] = c[r];      // cols 10..15 are exact zeros
    }
}

// ---------------- Per-node logits + init for layer 2 (D=10) --------------
__global__ void k_prep2(const float* __restrict__ h2, const float* __restrict__ a_src,
                        const float* __restrict__ a_dst, float* __restrict__ asrc,
                        float* __restrict__ adst, unsigned* __restrict__ mx,
                        float* __restrict__ denom, float* __restrict__ agg, int N) {
    int i = blockIdx.x * blockDim.x + threadIdx.x;
    if (i >= N) return;
    const float* hr = h2 + (size_t)i * 16;
    float s = 0.f, d = 0.f;
#pragma unroll
    for (int j = 0; j < 10; ++j) { float v = hr[j]; s += v * a_src[j]; d += v * a_dst[j]; }
    asrc[i] = s; adst[i] = d;
    mx[i] = 0u; denom[i] = 0.f;
    float* ar = agg + (size_t)i * 16;
#pragma unroll
    for (int j = 0; j < 16; ++j) ar[j] = 0.f;
}

// ---------------- MLP head: relu((agg2+b2)@Wl1+bl1) @ Wl2 + bl2 ----------
__global__ void k_final(const float* __restrict__ agg2, const float* __restrict__ b2,
                        const float* __restrict__ Wl1, const float* __restrict__ bl1,
                        const float* __restrict__ Wl2, const float* __restrict__ bl2,
                        float* __restrict__ out, int N) {
    int i = blockIdx.x * blockDim.x + threadIdx.x;
    if (i >= N) return;
    const float* ar = agg2 + (size_t)i * 16;
    float o[10];
#pragma unroll
    for (int j = 0; j < 10; ++j) o[j] = ar[j] + b2[j];
    float y = bl2[0];
#pragma unroll
    for (int k = 0; k < 10; ++k) {
        float t = bl1[k];
#pragma unroll
        for (int j = 0; j < 10; ++j) t += o[j] * Wl1[j * 10 + k];
        t = fmaxf(t, 0.f);
        y += t * Wl2[k];
    }
    out[i] = y;
}

// ------------------------------- launch ----------------------------------
extern "C" void kernel_launch(void* const* d_in, const int* in_sizes, int n_in,
                              void* d_out, int out_size, void* d_ws, size_t ws_size,
                              hipStream_t stream) {
    const float* x   = (const float*)d_in[0];
    const int*   ei  = (const int*)d_in[1];
    const float* W1  = (const float*)d_in[2];
    const float* a1s = (const float*)d_in[3];
    const float* a1d = (const float*)d_in[4];
    const float* b1  = (const float*)d_in[5];
    const float* W2  = (const float*)d_in[6];
    const float* a2s = (const float*)d_in[7];
    const float* a2d = (const float*)d_in[8];
    const float* b2  = (const float*)d_in[9];
    const float* Wl1 = (const float*)d_in[10];
    const float* bl1 = (const float*)d_in[11];
    const float* Wl2 = (const float*)d_in[12];
    const float* bl2 = (const float*)d_in[13];

    int N = in_sizes[0] / 128;
    int E = in_sizes[1] / 2;
    const int* src = ei;         // edge_index[0] = message sources
    const int* dst = ei + E;     // edge_index[1] = message targets

    // Workspace carve-up (~29 MB for N=100k), 256B-aligned slices.
    char* w = (char*)d_ws;
    auto alloc = [&](size_t bytes) {
        char* p = w;
        w += (bytes + 255) & ~(size_t)255;
        return (void*)p;
    };
    float*    h1   = (float*)alloc((size_t)N * 16 * 4);
    float*    agg1 = (float*)alloc((size_t)N * 16 * 4);
    float*    h2   = (float*)alloc((size_t)N * 16 * 4);
    float*    agg2 = (float*)alloc((size_t)N * 16 * 4);
    float*    as1  = (float*)alloc((size_t)N * 4);
    float*    ad1  = (float*)alloc((size_t)N * 4);
    float*    as2  = (float*)alloc((size_t)N * 4);
    float*    ad2  = (float*)alloc((size_t)N * 4);
    unsigned* mx1  = (unsigned*)alloc((size_t)N * 4);
    float*    dn1  = (float*)alloc((size_t)N * 4);
    unsigned* mx2  = (unsigned*)alloc((size_t)N * 4);
    float*    dn2  = (float*)alloc((size_t)N * 4);

    int tiles      = (N + 15) / 16;
    int gemmBlocks = (tiles + 7) / 8;         // 8 wave32 tiles per 256-thread block
    int nodeBlocks = (N + 255) / 256;
    int M          = E + N;                   // messages incl. self loops
    int edgeBlocks = (M + 255) / 256;

    // Layer 1
    k_gemm1<<<gemmBlocks, 256, 0, stream>>>(x, W1, h1, N);
    k_prep1<<<nodeBlocks, 256, 0, stream>>>(h1, a1s, a1d, as1, ad1, mx1, dn1, agg1, N);
    k_edge_max<<<edgeBlocks, 256, 0, stream>>>(src, dst, as1, ad1, mx1, E, N);
    k_edge_sum<<<edgeBlocks, 256, 0, stream>>>(src, dst, as1, ad1, mx1, dn1, E, N);
    k_edge_agg<16><<<edgeBlocks, 256, 0, stream>>>(src, dst, as1, ad1, mx1, dn1, h1, agg1, E, N);

    // Layer 2
    k_gemm2<<<gemmBlocks, 256, 0, stream>>>(agg1, b1, W2, h2, N);
    k_prep2<<<nodeBlocks, 256, 0, stream>>>(h2, a2s, a2d, as2, ad2, mx2, dn2, agg2, N);
    k_edge_max<<<edgeBlocks, 256, 0, stream>>>(src, dst, as2, ad2, mx2, E, N);
    k_edge_sum<<<edgeBlocks, 256, 0, stream>>>(src, dst, as2, ad2, mx2, dn2, E, N);
    k_edge_agg<10><<<edgeBlocks, 256, 0, stream>>>(src, dst, as2, ad2, mx2, dn2, h2, agg2, E, N);

    // Head
    k_final<<<nodeBlocks, 256, 0, stream>>>(agg2, b2, Wl1, bl1, Wl2, bl2, (float*)d_out, N);
}